// ODKL_Loss_80479097193024
// MI455X (gfx1250) — compile-verified
//
#include <hip/hip_runtime.h>
#include <math.h>

typedef __attribute__((ext_vector_type(2))) float v2f;
typedef __attribute__((ext_vector_type(8))) float v8f;

#define EPS_F 1e-8f

#define NBATCH        64
#define PER_BATCH     (512 * 512)                           // 262144 elems / batch
#define THREADS       256                                   // 8 waves of 32
#define ITERS         8                                     // float4 loads per thread
#define BLOCKS_PER_B  (PER_BATCH / (THREADS * ITERS * 4))   // = 32
#define TOTAL_BLOCKS  (NBATCH * BLOCKS_PER_B)               // = 2048

// Workspace layout:
//   float        partials[TOTAL_BLOCKS]        @ offset 0        (8 KiB)
//   unsigned int counts[NBATCH][6]             @ offset 8192     (1.5 KiB)
//     cumulative counts: (t_gt0, t_gt2, t_gt4, p_gt0, p_gt2, p_gt4)
//     bins derived in finalize: bin0 = gt0-gt2, bin1 = gt2-gt4, bin2 = gt4

__device__ __forceinline__ float loss_elem(float p, float t) {
  // truth < 1 -> (relu(p) - t)^2 ; else (p - t)^2
  float pe = (t < 1.0f) ? fmaxf(p, 0.0f) : p;
  float d  = pe - t;
  return d * d;
}

__global__ void odkl_zero_counts(unsigned int* counts) {
  unsigned int i = threadIdx.x;
  if (i < NBATCH * 6) counts[i] = 0u;
}

// Cumulative predicates: each is v_cmp + carry-in add (2 VALU), no cndmask.
#define GTCNT(v, g0, g2, g4) \
  g0 += ((v) > 0.0f);        \
  g2 += ((v) > 2.0f);        \
  g4 += ((v) > 4.0f);

__global__ __launch_bounds__(THREADS) void odkl_main(
    const float* __restrict__ predict, const float* __restrict__ truth,
    float* __restrict__ partials, unsigned int* __restrict__ counts) {
  const int batch = blockIdx.y;
  const int tid   = threadIdx.x;

  __shared__ float        wsum[THREADS / 32];
  __shared__ unsigned int wcnt[6];
  if (tid < 6) wcnt[tid] = 0u;

  const size_t base4 = (size_t)batch * (PER_BATCH / 4) +
                       (size_t)blockIdx.x * (THREADS * ITERS);
  const float4* __restrict__ p4 = (const float4*)predict + base4;
  const float4* __restrict__ t4 = (const float4*)truth   + base4;

  v2f ones; ones[0] = 1.0f; ones[1] = 1.0f;   // A = 16x4 all-ones
  v8f acc = {};                               // C/D 16x16 accumulator tile

  unsigned int tg0 = 0, tg2 = 0, tg4 = 0;     // truth cumulative counts
  unsigned int pg0 = 0, pg2 = 0, pg4 = 0;     // predict cumulative counts

#pragma unroll
  for (int j = 0; j < ITERS; ++j) {
    float4 p = p4[j * THREADS + tid];
    float4 t = t4[j * THREADS + tid];

    GTCNT(t.x, tg0, tg2, tg4); GTCNT(t.y, tg0, tg2, tg4);
    GTCNT(t.z, tg0, tg2, tg4); GTCNT(t.w, tg0, tg2, tg4);
    GTCNT(p.x, pg0, pg2, pg4); GTCNT(p.y, pg0, pg2, pg4);
    GTCNT(p.z, pg0, pg2, pg4); GTCNT(p.w, pg0, pg2, pg4);

    // Feed 64 loss values per wave per WMMA as the B (4x16) operand.
    // D = ones(16x4) x B(4x16) + C  => every B element lands in each of the
    // 16 rows once; the full-tile sum is 16x the sum of the B elements.
    v2f b0, b1;
    b0[0] = loss_elem(p.x, t.x); b0[1] = loss_elem(p.y, t.y);
    b1[0] = loss_elem(p.z, t.z); b1[1] = loss_elem(p.w, t.w);
    acc = __builtin_amdgcn_wmma_f32_16x16x4_f32(false, ones, false, b0,
                                                (short)0, acc, false, false);
    acc = __builtin_amdgcn_wmma_f32_16x16x4_f32(false, ones, false, b1,
                                                (short)0, acc, false, false);
  }

  // Per-lane share of the D tile (8 of 256 entries), then wave32 reduce.
  float lane = acc[0] + acc[1] + acc[2] + acc[3] +
               acc[4] + acc[5] + acc[6] + acc[7];
#pragma unroll
  for (int off = 16; off > 0; off >>= 1) {
    lane += __shfl_xor(lane, off, 32);
    tg0  += __shfl_xor(tg0, off, 32);
    tg2  += __shfl_xor(tg2, off, 32);
    tg4  += __shfl_xor(tg4, off, 32);
    pg0  += __shfl_xor(pg0, off, 32);
    pg2  += __shfl_xor(pg2, off, 32);
    pg4  += __shfl_xor(pg4, off, 32);
  }

  __syncthreads();  // wcnt zero-init visible
  const int wave = tid >> 5;
  const int lid  = tid & 31;
  if (lid == 0) {
    wsum[wave] = lane;
    atomicAdd(&wcnt[0], tg0); atomicAdd(&wcnt[1], tg2); atomicAdd(&wcnt[2], tg4);
    atomicAdd(&wcnt[3], pg0); atomicAdd(&wcnt[4], pg2); atomicAdd(&wcnt[5], pg4);
  }
  __syncthreads();

  if (tid == 0) {
    float s = 0.0f;
#pragma unroll
    for (int w = 0; w < THREADS / 32; ++w) s += wsum[w];
    partials[blockIdx.y * gridDim.x + blockIdx.x] = s;  // still carries x16
  }
  if (tid < 6) atomicAdd(&counts[batch * 6 + tid], wcnt[tid]);  // integer: deterministic
}

__global__ __launch_bounds__(256) void odkl_finalize(
    const float* __restrict__ partials, const unsigned int* __restrict__ counts,
    float* __restrict__ out) {
  __shared__ float red[256];
  __shared__ float klred[NBATCH];
  const int tid = threadIdx.x;

  // Fixed-order reduction of the 2048 block partials (deterministic).
  float s = 0.0f;
  for (int i = tid; i < TOTAL_BLOCKS; i += 256) s += partials[i];
  red[tid] = s;
  __syncthreads();
#pragma unroll
  for (int off = 128; off > 0; off >>= 1) {
    if (tid < off) red[tid] += red[tid + off];
    __syncthreads();
  }

  // Per-batch KL on threads 0..63.
  if (tid < NBATCH) {
    unsigned int tgt0 = counts[tid * 6 + 0];
    unsigned int tgt2 = counts[tid * 6 + 1];
    unsigned int tgt4 = counts[tid * 6 + 2];
    unsigned int pgt0 = counts[tid * 6 + 3];
    unsigned int pgt2 = counts[tid * 6 + 4];
    unsigned int pgt4 = counts[tid * 6 + 5];
    // Exact integer bin recovery from cumulative counts.
    float ct[3], cp[3];
    ct[0] = (float)(tgt0 - tgt2) + EPS_F;
    ct[1] = (float)(tgt2 - tgt4) + EPS_F;
    ct[2] = (float)(tgt4)        + EPS_F;
    cp[0] = (float)(pgt0 - pgt2) + EPS_F;
    cp[1] = (float)(pgt2 - pgt4) + EPS_F;
    cp[2] = (float)(pgt4)        + EPS_F;

    float mt = fmaxf(ct[0], fmaxf(ct[1], ct[2]));
    float mp = fmaxf(cp[0], fmaxf(cp[1], cp[2]));
    float et0 = expf(ct[0] - mt), et1 = expf(ct[1] - mt), et2 = expf(ct[2] - mt);
    float ep0 = expf(cp[0] - mp), ep1 = expf(cp[1] - mp), ep2 = expf(cp[2] - mp);
    float ist = 1.0f / (et0 + et1 + et2);
    float isp = 1.0f / (ep0 + ep1 + ep2);
    float pt0 = et0 * ist, pt1 = et1 * ist, pt2 = et2 * ist;
    float pp0 = ep0 * isp, pp1 = ep1 * isp, pp2 = ep2 * isp;
    float kl = pt0 * (logf(pt0 + EPS_F) - logf(pp0 + EPS_F)) +
               pt1 * (logf(pt1 + EPS_F) - logf(pp1 + EPS_F)) +
               pt2 * (logf(pt2 + EPS_F) - logf(pp2 + EPS_F));
    klred[tid] = kl;
  }
  __syncthreads();

  if (tid == 0) {
    float kls = 0.0f;
#pragma unroll
    for (int b = 0; b < NBATCH; ++b) kls += klred[b];
    // WMMA accumulator counted every element 16x: fold into the mean divisor.
    const float inv_den = 1.0f / (16.0f * (float)NBATCH * (float)PER_BATCH);
    float loss1   = red[0] * inv_den;
    float kl_mean = kls * (1.0f / (float)NBATCH);
    out[0] = 1.2f * loss1 + 2.0f * kl_mean;
  }
}

extern "C" void kernel_launch(void* const* d_in, const int* in_sizes, int n_in,
                              void* d_out, int out_size, void* d_ws, size_t ws_size,
                              hipStream_t stream) {
  (void)in_sizes; (void)n_in; (void)out_size; (void)ws_size;
  const float* predict = (const float*)d_in[0];
  const float* truth   = (const float*)d_in[1];

  float*        partials = (float*)d_ws;
  unsigned int* counts   = (unsigned int*)((char*)d_ws + TOTAL_BLOCKS * sizeof(float));

  odkl_zero_counts<<<1, 512, 0, stream>>>(counts);
  odkl_main<<<dim3(BLOCKS_PER_B, NBATCH), THREADS, 0, stream>>>(
      predict, truth, partials, counts);
  odkl_finalize<<<1, 256, 0, stream>>>(partials, counts, (float*)d_out);
}